// GatedGraphLayer_55027120996491
// MI455X (gfx1250) — compile-verified
//
#include <hip/hip_runtime.h>
#include <hip/hip_bf16.h>

#define DD 128   // channels

typedef float v8f __attribute__((ext_vector_type(8)));
typedef float v2f __attribute__((ext_vector_type(2)));
typedef unsigned int u32x4 __attribute__((ext_vector_type(4)));
typedef int i32x8 __attribute__((ext_vector_type(8)));
typedef int i32x4 __attribute__((ext_vector_type(4)));

#if defined(__has_builtin)
#if __has_builtin(__builtin_amdgcn_tensor_load_to_lds) && \
    __has_builtin(__builtin_amdgcn_s_wait_tensorcnt)
#define USE_TDM 1
#else
#define USE_TDM 0
#endif
#else
#define USE_TDM 0
#endif

// ---------------------------------------------------------------------------
// Wave-level 16x16 f32 GEMM tile: acc += A(16xK=128, LDS row-major) * B(K x 16)
// B read from global: btrans=1 -> B[k][n] = W[(colBase+n)*ldb + k]  (x @ W^T)
//                     btrans=0 -> B[k][n] = W[k*ldb + colBase+n]    (x @ W)
// VGPR layouts per CDNA5 ISA 7.12.2 (V_WMMA_F32_16X16X4_F32).
// ---------------------------------------------------------------------------
__device__ __forceinline__ v8f wave_gemm16(const float* __restrict__ A,
                                           const float* __restrict__ B,
                                           int ldb, bool btrans, int colBase,
                                           int lane, v8f acc) {
  const int hsel = lane >> 4;       // 0: K pair {0,1}, 1: K pair {2,3}
  const int nm   = lane & 15;       // A row / B column within tile
  const int n    = colBase + nm;
#pragma unroll
  for (int k0 = 0; k0 < DD; k0 += 4) {
    const int ka = k0 + 2 * hsel;
    v2f a, b;
    a[0] = A[nm * DD + ka];
    a[1] = A[nm * DD + ka + 1];
    if (btrans) {
      b[0] = B[n * ldb + ka];
      b[1] = B[n * ldb + ka + 1];
    } else {
      b[0] = B[ka * ldb + n];
      b[1] = B[(ka + 1) * ldb + n];
    }
    acc = __builtin_amdgcn_wmma_f32_16x16x4_f32(false, a, false, b,
                                                (short)0, acc, false, false);
  }
  return acc;
}

// C-tile element (i) sits at row = i + 8*hsel, col = colBase + (lane&15).
__device__ __forceinline__ void store_tile_lds(float* dst, v8f acc,
                                               int colBase, int lane) {
  const int hsel = lane >> 4, nm = lane & 15;
#pragma unroll
  for (int i = 0; i < 8; ++i) dst[(i + 8 * hsel) * DD + colBase + nm] = acc[i];
}

__device__ __forceinline__ void store_tile_global(float* __restrict__ dst,
                                                  int row0, int nrows, v8f acc,
                                                  float bv, int colBase, int lane) {
  const int hsel = lane >> 4, nm = lane & 15;
#pragma unroll
  for (int i = 0; i < 8; ++i) {
    const int r = row0 + i + 8 * hsel;
    if (r < nrows) dst[(size_t)r * DD + colBase + nm] = acc[i] + bv;
  }
}

// Fallback cooperative zero-padded 16x128 tile load into LDS (wave-local).
__device__ __forceinline__ void load_tile(float* lds, const float* __restrict__ g,
                                          int row0, int nrows, int lane) {
#pragma unroll
  for (int i = lane; i < 16 * (DD / 4); i += 32) {
    const int r = i >> 5, c4 = (i & 31) << 2;
    const int row = row0 + r;
    float4 v = make_float4(0.f, 0.f, 0.f, 0.f);
    if (row < nrows) v = *(const float4*)(g + (size_t)row * DD + c4);
    *(float4*)(lds + r * DD + c4) = v;
  }
}

// ---------------------------------------------------------------------------
// Tensor Data Mover: DMA one 16x128 f32 tile (rows row0..row0+15 of an
// [nrows x 128] row-major f32 tensor) from global into LDS at lds_off.
// D# bit layout per CDNA5 ISA 8.3/8.4:
//   G0: [1:0]=count=1, [63:32]=lds_addr, [120:64]=global_addr, [127:126]=type 2
//   G1: [17:16]=data_size(2->4B), dim0=128, dim1=rows_left (HW zero-fills OOB),
//       tile_dim0=128, tile_dim1=16, dim0_stride=128
// ---------------------------------------------------------------------------
__device__ __forceinline__ void tdm_load_tile_16x128(unsigned lds_off,
                                                     const float* __restrict__ g,
                                                     int row0, int nrows,
                                                     int lane) {
#if USE_TDM
  const unsigned long long ga =
      (unsigned long long)(const void*)(g + (size_t)row0 * DD);
  const unsigned rows_left = (unsigned)(nrows - row0);
  u32x4 g0;
  g0[0] = 1u;                                              // count=1, user D#
  g0[1] = lds_off;                                         // lds_addr (bytes)
  g0[2] = (unsigned)(ga & 0xFFFFFFFFull);                  // global_addr[31:0]
  g0[3] = (unsigned)((ga >> 32) & 0x1FFFFFFull) | (2u << 30); // addr[56:32]|type=2
  i32x8 g1;
  g1[0] = (int)(2u << 16);                                 // data_size = 4 bytes
  g1[1] = (int)(128u << 16);                               // tensor_dim0 = 128
  g1[2] = (int)((rows_left & 0xFFFFu) << 16);              // tensor_dim1 lo
  g1[3] = (int)((rows_left >> 16) | (128u << 16));         // dim1 hi | tile_dim0
  g1[4] = 16;                                              // tile_dim1=16
  g1[5] = 128;                                             // tensor_dim0_stride
  g1[6] = 0;
  g1[7] = 0;
  i32x4 zz4 = {0, 0, 0, 0};
  i32x8 zz8 = {0, 0, 0, 0, 0, 0, 0, 0};
  __builtin_amdgcn_tensor_load_to_lds(g0, g1, zz4, zz4, zz8, 0);
  (void)lane;
#else
  load_tile((float*)(size_t)0, g, row0, nrows, lane);  // unreachable fallback
#endif
}

__device__ __forceinline__ void tile_load_wait() {
#if USE_TDM
  __builtin_amdgcn_s_wait_tensorcnt(0);
#endif
  __builtin_amdgcn_wave_barrier();
}

// ---------------------------------------------------------------------------
// K1: h = x @ W_in^T + b_in ; m = h @ W_msg      (fused, h staged in LDS)
// ---------------------------------------------------------------------------
__global__ __launch_bounds__(64) void ggnn_proj_msg(
    const float* __restrict__ x, const float* __restrict__ Win,
    const float* __restrict__ bin, const float* __restrict__ Wmsg,
    float* __restrict__ h, float* __restrict__ m, int N) {
  __shared__ float sX[2][16 * DD];
  __shared__ float sH[2][16 * DD];
  const int wv = threadIdx.x >> 5, lane = threadIdx.x & 31;
  const int row0 = (blockIdx.x * 2 + wv) * 16;
  if (row0 >= N) return;                 // wave-uniform; no block barriers used
  float* ldsX = sX[wv];
  float* ldsH = sH[wv];

#if USE_TDM
  tdm_load_tile_16x128((unsigned)(size_t)ldsX, x, row0, N, lane);
#else
  load_tile(ldsX, x, row0, N, lane);
#endif
  tile_load_wait();

  const int nm = lane & 15, hsel = lane >> 4;
#pragma unroll
  for (int ct = 0; ct < 8; ++ct) {
    const int cb = ct * 16;
    v8f acc = {0, 0, 0, 0, 0, 0, 0, 0};
    acc = wave_gemm16(ldsX, Win, DD, true, cb, lane, acc);
    const float bv = bin[cb + nm];
#pragma unroll
    for (int i = 0; i < 8; ++i) acc[i] += bv;
    store_tile_lds(ldsH, acc, cb, lane);
#pragma unroll
    for (int i = 0; i < 8; ++i) {
      const int r = row0 + i + 8 * hsel;
      if (r < N) h[(size_t)r * DD + cb + nm] = acc[i];
    }
  }
  __builtin_amdgcn_wave_barrier();
#pragma unroll
  for (int ct = 0; ct < 8; ++ct) {
    v8f acc = {0, 0, 0, 0, 0, 0, 0, 0};
    acc = wave_gemm16(ldsH, Wmsg, DD, false, ct * 16, lane, acc);
    store_tile_global(m, row0, N, acc, 0.f, ct * 16, lane);
  }
}

// ---------------------------------------------------------------------------
// K2: zero agg
// ---------------------------------------------------------------------------
__global__ void ggnn_zero(float4* __restrict__ p, long long n4) {
  const long long i = (long long)blockIdx.x * blockDim.x + threadIdx.x;
  if (i < n4) p[i] = make_float4(0.f, 0.f, 0.f, 0.f);
}

// ---------------------------------------------------------------------------
// K3: agg[dst] += m[src]   (one thread per (edge, 4-channel chunk))
// ---------------------------------------------------------------------------
__global__ void ggnn_scatter(const float* __restrict__ m,
                             const int* __restrict__ ei,
                             float* __restrict__ agg, int E) {
  const long long t = (long long)blockIdx.x * blockDim.x + threadIdx.x;
  const int e = (int)(t >> 5);
  if (e >= E) return;
  const int c = ((int)t & 31) << 2;
  const int s = ei[e];
  const int d = ei[(size_t)E + e];
  const float4 v = *(const float4*)(m + (size_t)s * DD + c);
  float* a = agg + (size_t)d * DD + c;
  atomicAdd(a + 0, v.x);
  atomicAdd(a + 1, v.y);
  atomicAdd(a + 2, v.z);
  atomicAdd(a + 3, v.w);
}

// ---------------------------------------------------------------------------
// K4: fused GRU + output projection
//   gi = agg @ W_ih^T + b_ih ; gh = h @ W_hh^T + b_hh (per 16-col tile)
//   gate math in registers; h_new staged in LDS; out = h_new @ W_out^T + b_out
// ---------------------------------------------------------------------------
__global__ __launch_bounds__(64) void ggnn_gru_out(
    const float* __restrict__ agg, const float* __restrict__ h,
    const float* __restrict__ Wih, const float* __restrict__ bih,
    const float* __restrict__ Whh, const float* __restrict__ bhh,
    const float* __restrict__ Wout, const float* __restrict__ bout,
    float* __restrict__ out, int N) {
  __shared__ float sA[2][16 * DD];
  __shared__ float sH[2][16 * DD];
  __shared__ float sO[2][16 * DD];
  const int wv = threadIdx.x >> 5, lane = threadIdx.x & 31;
  const int row0 = (blockIdx.x * 2 + wv) * 16;
  if (row0 >= N) return;
  float* ldsA = sA[wv];
  float* ldsH = sH[wv];
  float* ldsO = sO[wv];

#if USE_TDM
  tdm_load_tile_16x128((unsigned)(size_t)ldsA, agg, row0, N, lane);
  tdm_load_tile_16x128((unsigned)(size_t)ldsH, h, row0, N, lane);
#else
  load_tile(ldsA, agg, row0, N, lane);
  load_tile(ldsH, h, row0, N, lane);
#endif
  tile_load_wait();

  const int nm = lane & 15, hsel = lane >> 4;
  for (int ct = 0; ct < 8; ++ct) {
    const int cb = ct * 16;
    v8f zero = {0, 0, 0, 0, 0, 0, 0, 0};
    v8f ir  = wave_gemm16(ldsA, Wih, DD, true, cb, lane, zero);
    v8f iz  = wave_gemm16(ldsA, Wih, DD, true, cb + DD, lane, zero);
    v8f in_ = wave_gemm16(ldsA, Wih, DD, true, cb + 2 * DD, lane, zero);
    v8f hr  = wave_gemm16(ldsH, Whh, DD, true, cb, lane, zero);
    v8f hz  = wave_gemm16(ldsH, Whh, DD, true, cb + DD, lane, zero);
    v8f hn  = wave_gemm16(ldsH, Whh, DD, true, cb + 2 * DD, lane, zero);
    const float bir = bih[cb + nm], biz = bih[DD + cb + nm], bin_ = bih[2 * DD + cb + nm];
    const float bhr = bhh[cb + nm], bhz = bhh[DD + cb + nm], bhn = bhh[2 * DD + cb + nm];
#pragma unroll
    for (int i = 0; i < 8; ++i) {
      const float r = 1.f / (1.f + __expf(-(ir[i] + bir + hr[i] + bhr)));
      const float z = 1.f / (1.f + __expf(-(iz[i] + biz + hz[i] + bhz)));
      const float nn = tanhf(in_[i] + bin_ + r * (hn[i] + bhn));
      const float hv = ldsH[(i + 8 * hsel) * DD + cb + nm];
      ldsO[(i + 8 * hsel) * DD + cb + nm] = (1.f - z) * nn + z * hv;
    }
  }
  __builtin_amdgcn_wave_barrier();
#pragma unroll
  for (int ct = 0; ct < 8; ++ct) {
    v8f acc = {0, 0, 0, 0, 0, 0, 0, 0};
    acc = wave_gemm16(ldsO, Wout, DD, true, ct * 16, lane, acc);
    store_tile_global(out, row0, N, acc, bout[ct * 16 + nm], ct * 16, lane);
  }
}

// ---------------------------------------------------------------------------
extern "C" void kernel_launch(void* const* d_in, const int* in_sizes, int n_in,
                              void* d_out, int out_size, void* d_ws, size_t ws_size,
                              hipStream_t stream) {
  const float* x    = (const float*)d_in[0];
  const int*   ei   = (const int*)d_in[1];
  const float* Win  = (const float*)d_in[2];
  const float* bin  = (const float*)d_in[3];
  const float* Wmsg = (const float*)d_in[4];
  const float* Wih  = (const float*)d_in[5];
  const float* bih  = (const float*)d_in[6];
  const float* Whh  = (const float*)d_in[7];
  const float* bhh  = (const float*)d_in[8];
  const float* Wout = (const float*)d_in[9];
  const float* bout = (const float*)d_in[10];

  const int N = in_sizes[0] / DD;
  const int E = in_sizes[1] / 2;

  float* h   = (float*)d_ws;
  float* m   = h + (size_t)N * DD;
  float* agg = m + (size_t)N * DD;
  float* o   = (float*)d_out;

  const int tiles  = (N + 15) / 16;
  const int blocks = (tiles + 1) / 2;   // 2 waves (tiles) per 64-thread block

  ggnn_proj_msg<<<blocks, 64, 0, stream>>>(x, Win, bin, Wmsg, h, m, N);

  const long long n4 = (long long)N * DD / 4;
  ggnn_zero<<<(int)((n4 + 255) / 256), 256, 0, stream>>>((float4*)agg, n4);

  const long long st = (long long)E * 32;
  ggnn_scatter<<<(int)((st + 255) / 256), 256, 0, stream>>>(m, ei, agg, E);

  ggnn_gru_out<<<blocks, 64, 0, stream>>>(agg, h, Wih, bih, Whh, bhh,
                                          Wout, bout, o, N);
}